// PointsTimes_25383256719963
// MI455X (gfx1250) — compile-verified
//
#include <hip/hip_runtime.h>
#include <stdint.h>

// Problem constants (match reference)
#define CCH   160   // channels
#define NPTS  500   // points
#define NP    8     // neighbors per point
#define NIDX  (NPTS * NP)   // 4000 int64 indices

// Tiling: each block = 8 channels x 125 points (exact: 4*125 = 500)
#define CBLK    8
#define NBLK    125
#define THREADS 256         // 8 wave32; wave w owns channel c0+w

typedef uint32_t v4u __attribute__((ext_vector_type(4)));
typedef int      v8i __attribute__((ext_vector_type(8)));
typedef int      v4i __attribute__((ext_vector_type(4)));

__global__ __launch_bounds__(THREADS)
void points_times_gather_kernel(const float* __restrict__ feat1,
                                const float* __restrict__ feat2,
                                const long long* __restrict__ inds,
                                float* __restrict__ out) {
    __shared__ float s_rows[CBLK * NPTS];   // 16000 B: 8 contiguous feat2 rows
    __shared__ int   s_idx[NBLK * NP];      // 4000 B: int32 neighbor indices

    const int n0   = blockIdx.x * NBLK;     // point-tile origin (0,125,250,375)
    const int c0   = blockIdx.y * CBLK;     // channel-tile origin
    const int tid  = threadIdx.x;
    const int lane = tid & 31;
    const int wave = tid >> 5;              // 0..7 == local channel

    // ---- Stage indices: threads 0..249 own 4 consecutive int64 each ----
    // All in-bounds by construction: g <= 3996. int64->int32 = low dword (LE).
    if (tid < (NBLK * NP) / 4) {            // tid < 250
        const int g = n0 * NP + 4 * tid;    // first int64 element this thread owns
        const int4* p4 = (const int4*)inds; // one int4 == two int64
        const int4 a = p4[(g >> 1) + 0];    // global_load_b128
        const int4 b = p4[(g >> 1) + 1];    // global_load_b128
        int4 v; v.x = a.x; v.y = a.z; v.z = b.x; v.w = b.z;
        *(int4*)&s_idx[4 * tid] = v;        // ds_store_b128
    }

#if __has_builtin(__builtin_amdgcn_tensor_load_to_lds)
    // ---- TDM: one async DMA of the contiguous 16000B row block -> LDS ----
    if (tid == 0) {
        const uint64_t gaddr = (uint64_t)(uintptr_t)(feat2 + (size_t)c0 * NPTS);
        const uint32_t laddr = (uint32_t)(uintptr_t)(const void*)s_rows; // LDS byte addr
        const uint32_t nelem = CBLK * NPTS;  // 4000 f32 elements

        // D# group 0 (128b): count=1 | lds_addr | global_addr[56:0] | type=2
        v4u g0 = { 1u,
                   laddr,
                   (uint32_t)gaddr,
                   (uint32_t)((gaddr >> 32) & 0x1FFFFFFu) | (2u << 30) };
        // D# group 1 (256b): data_size=4B; tensor_dim0=4000 (bits 79:48);
        // tensor_dim1=1 (bits 111:80); tile_dim0=4000 (bits 127:112);
        // tile_dim1=1 (bits 143:128); tensor_dim0_stride=4000 (bits 207:160)
        v8i g1 = { (int)(2u << 16),
                   (int)((nelem & 0xFFFFu) << 16),
                   (int)(((nelem >> 16) & 0xFFFFu) | (1u << 16)),
                   (int)((nelem & 0xFFFFu) << 16),
                   1,
                   (int)nelem,
                   0, 0 };
        v4i z4 = { 0, 0, 0, 0 };
#if defined(__clang_major__) && (__clang_major__ >= 23)
        v8i z8 = { 0, 0, 0, 0, 0, 0, 0, 0 };
        __builtin_amdgcn_tensor_load_to_lds(g0, g1, z4, z4, z8, 0);
#else
        __builtin_amdgcn_tensor_load_to_lds(g0, g1, z4, z4, 0);
#endif
    }
    // All waves execute: issuing wave drains TENSORcnt, others trivially pass.
    __builtin_amdgcn_s_wait_tensorcnt(0);
#else
    // Fallback: cooperative vectorized copy of the contiguous 16000B block.
    {
        const float4* src = (const float4*)(feat2 + (size_t)c0 * NPTS);
        float4* dst = (float4*)s_rows;
        for (int i = tid; i < (CBLK * NPTS) / 4; i += THREADS) dst[i] = src[i];
    }
#endif
    __syncthreads();

    // ---- Compute: wave w == channel c0+w; lanes stride the 125 points ----
    {
        const float* row   = &s_rows[wave * NPTS];
        const int    c     = c0 + wave;
        const float* f1row = feat1 + (size_t)c * NPTS + n0;
        float*       orow  = out   + (size_t)c * NPTS + n0;

        #pragma unroll
        for (int k = 0; k < 4; ++k) {
            const int nl = lane + 32 * k;           // k<=2 provably < 125
            if (nl < NBLK) {
                const int4 i0 = *(const int4*)&s_idx[nl * NP + 0];  // ds_load_b128
                const int4 i1 = *(const int4*)&s_idx[nl * NP + 4];  // ds_load_b128
                float s = row[i0.x] + row[i0.y] + row[i0.z] + row[i0.w]
                        + row[i1.x] + row[i1.y] + row[i1.z] + row[i1.w];
                orow[nl] = f1row[nl] * s * 0.125f;
            }
        }
    }
}

extern "C" void kernel_launch(void* const* d_in, const int* in_sizes, int n_in,
                              void* d_out, int out_size, void* d_ws, size_t ws_size,
                              hipStream_t stream) {
    const float*     feat1 = (const float*)d_in[0];
    const float*     feat2 = (const float*)d_in[1];
    const long long* inds  = (const long long*)d_in[2];
    float*           out   = (float*)d_out;

    dim3 grid(NPTS / NBLK, CCH / CBLK);   // (4, 20) = 80 blocks
    points_times_gather_kernel<<<grid, THREADS, 0, stream>>>(feat1, feat2, inds, out);
}